// GOEnricher_19628000542883
// MI455X (gfx1250) — compile-verified
//
#include <hip/hip_runtime.h>

typedef __attribute__((ext_vector_type(2))) float v2f;
typedef __attribute__((ext_vector_type(8))) float v8f;
typedef __attribute__((ext_vector_type(4))) unsigned int v4u;
typedef __attribute__((ext_vector_type(8))) int v8i;
typedef __attribute__((ext_vector_type(4))) int v4i;

#define N_PROT 50000
#define N_GO   50000
#define N_EDGE 1600000
#define GO_DIM 200
#define HID    256
#define LN_EPS 1e-5f

#define SP_STRIDE 258   // prot tile stride (TDM pad: 2 DWORDs per 256-DWORD row)
#define SG_STRIDE 257   // agg / T tile stride (odd -> bank friendly)

// Workspace layout (bytes, 16B aligned)
#define WS_GOH      0            // 50000*256*4 = 51,200,000
#define WS_COUNTS   51200000     // 50000*4    =    200,000
#define WS_OFFSETS  51400000     // 50001*4    =    200,004
#define WS_CURSOR   51600016     // 50000*4    =    200,000
#define WS_SORTED   51800016     // 1600000*4  =  6,400,000

// ---------------- CSR build ----------------

__global__ void k_zero_counts(int* __restrict__ counts) {
    int i = blockIdx.x * blockDim.x + threadIdx.x;
    if (i < N_PROT) counts[i] = 0;
}

__global__ void k_hist(const int* __restrict__ edges, int* __restrict__ counts) {
    int e = blockIdx.x * blockDim.x + threadIdx.x;
    if (e < N_EDGE) atomicAdd(&counts[edges[e]], 1);  // edges[0:E] = dst
}

// single block, 1024 threads: exclusive scan of counts -> offsets, cursor
__global__ void __launch_bounds__(1024)
k_scan(const int* __restrict__ counts, int* __restrict__ offsets,
       int* __restrict__ cursor) {
    __shared__ int part[1024];
    int t = threadIdx.x;
    const int chunk = (N_PROT + 1023) / 1024;
    int b = t * chunk;
    int e = b + chunk; if (e > N_PROT) e = N_PROT;
    int s = 0;
    for (int i = b; i < e && b < N_PROT; ++i) s += counts[i];
    part[t] = s;
    __syncthreads();
    // Hillis-Steele inclusive scan
    for (int off = 1; off < 1024; off <<= 1) {
        int v = (t >= off) ? part[t - off] : 0;
        __syncthreads();
        part[t] += v;
        __syncthreads();
    }
    int run = part[t] - s;  // exclusive prefix of this chunk
    for (int i = b; i < e && b < N_PROT; ++i) {
        offsets[i] = run;
        cursor[i]  = run;
        run += counts[i];
    }
    if (t == 1023) offsets[N_PROT] = part[1023];
}

__global__ void k_scatter(const int* __restrict__ edges, int* __restrict__ cursor,
                          int* __restrict__ sorted_src) {
    int e = blockIdx.x * blockDim.x + threadIdx.x;
    if (e < N_EDGE) {
        int d = edges[e];
        int s = edges[N_EDGE + e];
        int pos = atomicAdd(&cursor[d], 1);
        sorted_src[pos] = s;
    }
}

// ---------------- GEMM 1: go_h = relu(go_x @ Wg + bg) ----------------
// 512 threads = 16 waves; block handles 16 rows; wave w owns 16-col tile w.
// V_WMMA_F32_16X16X4_F32, K = 200 -> 50 steps.

__global__ void __launch_bounds__(512)
k_goproj(const float* __restrict__ go_x, const float* __restrict__ Wg,
         const float* __restrict__ bg, float* __restrict__ go_h) {
    __shared__ float shA[16 * 201];  // stride 201 (odd mod 64) -> conflict-free
    const int tid  = threadIdx.x;
    const int row0 = blockIdx.x * 16;

    for (int i = tid; i < 16 * GO_DIM; i += 512) {
        int r = i / GO_DIM, k = i - r * GO_DIM;
        shA[r * 201 + k] = go_x[(row0 + r) * GO_DIM + k];
    }
    __syncthreads();

    const int wave = tid >> 5;
    const int lane = tid & 31;
    const int m    = lane & 15;       // row (A) / col (B,D)
    const int hi   = lane >> 4;
    const int koff = hi * 2;
    const int c0   = wave * 16;

    v8f acc = {};
    for (int k0 = 0; k0 < GO_DIM; k0 += 4) {
        v2f a, bfrag;
        a.x     = shA[m * 201 + k0 + koff];
        a.y     = shA[m * 201 + k0 + koff + 1];
        bfrag.x = Wg[(k0 + koff) * HID + c0 + m];
        bfrag.y = Wg[(k0 + koff + 1) * HID + c0 + m];
        acc = __builtin_amdgcn_wmma_f32_16x16x4_f32(
            false, a, false, bfrag, (short)0, acc, false, false);
    }

    const float bias = bg[c0 + m];
#pragma unroll
    for (int j = 0; j < 8; ++j) {
        int mr = j + 8 * hi;
        float v = acc[j] + bias;
        v = v > 0.f ? v : 0.f;
        go_h[(row0 + mr) * HID + c0 + m] = v;
    }
}

// ---------------- Fused: TDM prot prefetch + aggregate + MLP + LN ----------
// 256 threads = 8 waves; block handles 16 proteins; wave w owns col tiles w, w+8.
// The 16x256 prot tile is DMA'd into LDS by the Tensor Data Mover, overlapped
// with the CSR aggregation loop; wave 0 waits TENSORcnt and the barrier
// publishes the tile to all waves.

__global__ void __launch_bounds__(256)
k_fuse(const float* __restrict__ prot, const float* __restrict__ go_h,
       const int* __restrict__ offsets, const int* __restrict__ sorted_src,
       const float* __restrict__ W1, const float* __restrict__ b1,
       const float* __restrict__ W2, const float* __restrict__ b2,
       const float* __restrict__ gamma, const float* __restrict__ beta,
       float* __restrict__ out) {
    __shared__ float shP[16 * SP_STRIDE];  // prot tile (TDM, 2-DW row pad)
    __shared__ float shG[16 * SG_STRIDE];  // go_agg, later reused for F
    __shared__ float shT[16 * SG_STRIDE];  // relu(H@W1+b1)
    __shared__ float presf[16];

    const int tid = threadIdx.x;
    const int p0  = blockIdx.x * 16;

    // ---- TDM: async DMA prot[p0:p0+16, 0:256] -> shP (wave 0 only) ----
    if (tid < 32) {
        unsigned long long ga = (unsigned long long)(const void*)(prot + (size_t)p0 * HID);
        unsigned lds_off = (unsigned)(size_t)(const void*)&shP[0];
        v4u g0;
        g0[0] = 1u;                                  // count=1, user descriptor
        g0[1] = lds_off;                             // lds_addr (bytes)
        g0[2] = (unsigned)(ga & 0xFFFFFFFFu);        // global_addr[31:0]
        g0[3] = (unsigned)((ga >> 32) & 0x01FFFFFFu) // global_addr[56:32]
                | (2u << 30);                        // type=2 (image)
        v8i g1;
        g1[0] = (int)((2u << 16)      // data_size = 4B
                    | (1u << 20)      // pad_enable
                    | (7u << 22)      // pad_interval: every 256 DWORDs
                    | (1u << 25));    // pad_amount: 2 DWORDs -> stride 258
        g1[1] = (int)(256u << 16);    // tensor_dim0[15:0]=256
        g1[2] = (int)(16u << 16);     // tensor_dim0 hi=0 | tensor_dim1[15:0]=16
        g1[3] = (int)(256u << 16);    // tensor_dim1 hi=0 | tile_dim0=256
        g1[4] = 16;                   // tile_dim1=16, tile_dim2=0
        g1[5] = 256;                  // tensor_dim0_stride[31:0]=256
        g1[6] = 0;
        g1[7] = 0;
        v4i gz4 = {0, 0, 0, 0};
        v8i gz8 = {0, 0, 0, 0, 0, 0, 0, 0};
        __builtin_amdgcn_tensor_load_to_lds(g0, g1, gz4, gz4, gz8, 0);
    }

    // ---- CSR aggregation (overlaps the TDM DMA) ----
    for (int r = 0; r < 16; ++r) {
        int p   = p0 + r;
        int beg = offsets[p], end = offsets[p + 1];
        float acc = 0.f;
        for (int e = beg; e < end; ++e) {
            int s = sorted_src[e];            // uniform (scalar) load
            acc += go_h[s * HID + tid];       // coalesced 1KB row from L2
        }
        int cnt = end - beg;
        float denom = cnt > 0 ? (float)cnt : 1.f;
        shG[r * SG_STRIDE + tid] = acc / denom;
        if (tid == 0) presf[r] = cnt > 0 ? 1.f : 0.f;
    }

    if (tid < 32) __builtin_amdgcn_s_wait_tensorcnt(0);
    __syncthreads();

    const int wave = tid >> 5;
    const int lane = tid & 31;
    const int m    = lane & 15;
    const int hi   = lane >> 4;
    const int koff = hi * 2;
    const int c0a  = wave * 16;
    const int c0b  = (wave + 8) * 16;

    // GEMM1: T = relu([prot|agg] @ W1 + b1), K = 512 as two K=256 halves
    v8f t0 = {}, t1 = {};
    for (int k0 = 0; k0 < HID; k0 += 4) {        // first half: A = prot (shP)
        v2f a, ba, bb;
        a.x  = shP[m * SP_STRIDE + k0 + koff];
        a.y  = shP[m * SP_STRIDE + k0 + koff + 1];
        ba.x = W1[(k0 + koff) * HID + c0a + m];
        ba.y = W1[(k0 + koff + 1) * HID + c0a + m];
        bb.x = W1[(k0 + koff) * HID + c0b + m];
        bb.y = W1[(k0 + koff + 1) * HID + c0b + m];
        t0 = __builtin_amdgcn_wmma_f32_16x16x4_f32(false, a, false, ba, (short)0, t0, false, false);
        t1 = __builtin_amdgcn_wmma_f32_16x16x4_f32(false, a, false, bb, (short)0, t1, false, false);
    }
    for (int k0 = 0; k0 < HID; k0 += 4) {        // second half: A = agg (shG)
        v2f a, ba, bb;
        a.x  = shG[m * SG_STRIDE + k0 + koff];
        a.y  = shG[m * SG_STRIDE + k0 + koff + 1];
        ba.x = W1[(HID + k0 + koff) * HID + c0a + m];
        ba.y = W1[(HID + k0 + koff + 1) * HID + c0a + m];
        bb.x = W1[(HID + k0 + koff) * HID + c0b + m];
        bb.y = W1[(HID + k0 + koff + 1) * HID + c0b + m];
        t0 = __builtin_amdgcn_wmma_f32_16x16x4_f32(false, a, false, ba, (short)0, t0, false, false);
        t1 = __builtin_amdgcn_wmma_f32_16x16x4_f32(false, a, false, bb, (short)0, t1, false, false);
    }
    {
        float biasA = b1[c0a + m], biasB = b1[c0b + m];
#pragma unroll
        for (int j = 0; j < 8; ++j) {
            int mr = j + 8 * hi;
            float va = t0[j] + biasA; va = va > 0.f ? va : 0.f;
            float vb = t1[j] + biasB; vb = vb > 0.f ? vb : 0.f;
            shT[mr * SG_STRIDE + c0a + m] = va;
            shT[mr * SG_STRIDE + c0b + m] = vb;
        }
    }
    __syncthreads();   // shT complete; shG reads finished (now reusable for F)

    // GEMM2: F = T @ W2 + b2, K = 256
    v8f f0 = {}, f1 = {};
    for (int k0 = 0; k0 < HID; k0 += 4) {
        v2f a, ba, bb;
        a.x  = shT[m * SG_STRIDE + k0 + koff];
        a.y  = shT[m * SG_STRIDE + k0 + koff + 1];
        ba.x = W2[(k0 + koff) * HID + c0a + m];
        ba.y = W2[(k0 + koff + 1) * HID + c0a + m];
        bb.x = W2[(k0 + koff) * HID + c0b + m];
        bb.y = W2[(k0 + koff + 1) * HID + c0b + m];
        f0 = __builtin_amdgcn_wmma_f32_16x16x4_f32(false, a, false, ba, (short)0, f0, false, false);
        f1 = __builtin_amdgcn_wmma_f32_16x16x4_f32(false, a, false, bb, (short)0, f1, false, false);
    }
    {
        // store F (+bias) into shG (free after GEMM1)
        float biasA = b2[c0a + m], biasB = b2[c0b + m];
#pragma unroll
        for (int j = 0; j < 8; ++j) {
            int mr = j + 8 * hi;
            shG[mr * SG_STRIDE + c0a + m] = f0[j] + biasA;
            shG[mr * SG_STRIDE + c0b + m] = f1[j] + biasB;
        }
    }
    __syncthreads();

    // Residual + LayerNorm; wave w handles rows w and w+8 (wave32 shuffles)
    for (int rr = 0; rr < 2; ++rr) {
        int r = wave + rr * 8;
        float x[8];
        float sum = 0.f, sumsq = 0.f;
#pragma unroll
        for (int j = 0; j < 8; ++j) {
            int col = lane + 32 * j;
            float v = shP[r * SP_STRIDE + col] + presf[r] * shG[r * SG_STRIDE + col];
            x[j] = v;
            sum += v;
            sumsq += v * v;
        }
        for (int off = 16; off >= 1; off >>= 1) {
            sum   += __shfl_xor(sum, off, 32);
            sumsq += __shfl_xor(sumsq, off, 32);
        }
        float mu   = sum * (1.f / HID);
        float var  = sumsq * (1.f / HID) - mu * mu;
        float rstd = rsqrtf(var + LN_EPS);
#pragma unroll
        for (int j = 0; j < 8; ++j) {
            int col = lane + 32 * j;
            out[(p0 + r) * HID + col] = (x[j] - mu) * rstd * gamma[col] + beta[col];
        }
    }
}

// ---------------- launcher ----------------

extern "C" void kernel_launch(void* const* d_in, const int* in_sizes, int n_in,
                              void* d_out, int out_size, void* d_ws, size_t ws_size,
                              hipStream_t stream) {
    const float* prot  = (const float*)d_in[0];
    const float* go_x  = (const float*)d_in[1];
    const int*   edges = (const int*)d_in[2];
    // d_in[3] = num_proteins scalar (compile-time constant here)
    const float* Wg    = (const float*)d_in[4];
    const float* bg    = (const float*)d_in[5];
    const float* W1    = (const float*)d_in[6];
    const float* b1    = (const float*)d_in[7];
    const float* W2    = (const float*)d_in[8];
    const float* b2    = (const float*)d_in[9];
    const float* gamma = (const float*)d_in[10];
    const float* beta  = (const float*)d_in[11];
    float* out = (float*)d_out;

    char* ws = (char*)d_ws;
    float* go_h    = (float*)(ws + WS_GOH);
    int*   counts  = (int*)(ws + WS_COUNTS);
    int*   offsets = (int*)(ws + WS_OFFSETS);
    int*   cursor  = (int*)(ws + WS_CURSOR);
    int*   sorted  = (int*)(ws + WS_SORTED);

    k_zero_counts<<<(N_PROT + 255) / 256, 256, 0, stream>>>(counts);
    k_hist<<<(N_EDGE + 255) / 256, 256, 0, stream>>>(edges, counts);
    k_scan<<<1, 1024, 0, stream>>>(counts, offsets, cursor);
    k_scatter<<<(N_EDGE + 255) / 256, 256, 0, stream>>>(edges, cursor, sorted);
    k_goproj<<<N_GO / 16, 512, 0, stream>>>(go_x, Wg, bg, go_h);
    k_fuse<<<N_PROT / 16, 256, 0, stream>>>(prot, go_h, offsets, sorted,
                                            W1, b1, W2, b2, gamma, beta, out);
}